// GNN_PDE_1279900254338
// MI455X (gfx1250) — compile-verified
//
#include <hip/hip_runtime.h>
#include <hip/hip_bf16.h>

typedef __attribute__((ext_vector_type(2))) float v2f;
typedef __attribute__((ext_vector_type(8))) float v8f;

// ---------------------------------------------------------------------------
// Degree / normalization kernels
// ---------------------------------------------------------------------------
__global__ void deg_init_kernel(float* deg, int n) {
    int i = blockIdx.x * blockDim.x + threadIdx.x;
    if (i < n) deg[i] = 1.0f;  // self-loop contributes 1 to every node's degree
}

__global__ void deg_count_kernel(const long long* __restrict__ dst, float* deg, int e) {
    int i = blockIdx.x * blockDim.x + threadIdx.x;
    if (i < e) atomicAdd(&deg[dst[i]], 1.0f);
}

__global__ void deg_finalize_kernel(float* dinv, int n) {
    int i = blockIdx.x * blockDim.x + threadIdx.x;
    if (i < n) dinv[i] = rsqrtf(dinv[i]);  // deg >= 1 always (self loops)
}

// ---------------------------------------------------------------------------
// Dense transform Y[n_rows,64] = X[n_rows,K] @ W[K,64] using fp32 WMMA.
// Block = 128 threads = 4 waves; each wave computes a 16-row x 64-col slab.
// W is staged in LDS (K*64 floats, <= 32 KB << 320 KB/WGP).
//
// A-fragment layout (ISA 32-bit A 16x4):   lane L: M = L%16, holds K = 2*(L/16)+{0,1}
// B-fragment layout (row striped / lanes): lane L: N = L%16, holds K = 2*(L/16)+{0,1}
// C/D layout (32-bit 16x16): VGPR v, lane L: M = v + 8*(L/16), N = L%16
// ---------------------------------------------------------------------------
__global__ __launch_bounds__(128) void gemm_wmma_kernel(
    const float* __restrict__ X, const float* __restrict__ W,
    float* __restrict__ Y, int n_rows, int K) {
    __shared__ float Wlds[128 * 64];

    const int tid = threadIdx.x;
    for (int i = tid; i < K * 64; i += 128) Wlds[i] = W[i];
    __syncthreads();

    const int wave = tid >> 5;
    const int lane = tid & 31;
    const int m    = lane & 15;          // M (for A) / N-in-tile (for B,C)
    const int kh   = (lane >> 4) << 1;   // K sub-offset: 0 or 2
    const int rowBase = blockIdx.x * 64 + wave * 16;
    const int row  = rowBase + m;
    const bool rv  = row < n_rows;
    const float* xrow = X + (size_t)(rv ? row : 0) * (size_t)K;

    v8f acc0 = {}, acc1 = {}, acc2 = {}, acc3 = {};

    for (int k = 0; k < K; k += 4) {
        v2f a;
        float a0 = xrow[k + kh];
        float a1 = xrow[k + kh + 1];
        a.x = rv ? a0 : 0.0f;
        a.y = rv ? a1 : 0.0f;

        const int kr0 = (k + kh) * 64;
        const int kr1 = (k + kh + 1) * 64;
        v2f b0, b1, b2, b3;
        b0.x = Wlds[kr0 + m];       b0.y = Wlds[kr1 + m];
        b1.x = Wlds[kr0 + 16 + m];  b1.y = Wlds[kr1 + 16 + m];
        b2.x = Wlds[kr0 + 32 + m];  b2.y = Wlds[kr1 + 32 + m];
        b3.x = Wlds[kr0 + 48 + m];  b3.y = Wlds[kr1 + 48 + m];

        acc0 = __builtin_amdgcn_wmma_f32_16x16x4_f32(false, a, false, b0, (short)0, acc0, false, false);
        acc1 = __builtin_amdgcn_wmma_f32_16x16x4_f32(false, a, false, b1, (short)0, acc1, false, false);
        acc2 = __builtin_amdgcn_wmma_f32_16x16x4_f32(false, a, false, b2, (short)0, acc2, false, false);
        acc3 = __builtin_amdgcn_wmma_f32_16x16x4_f32(false, a, false, b3, (short)0, acc3, false, false);
    }

    const int rtop = (lane >> 4) << 3;  // 0 or 8
    #pragma unroll
    for (int v = 0; v < 8; ++v) {
        int r = rowBase + rtop + v;
        if (r < n_rows) {
            float* yr = Y + (size_t)r * 64;
            yr[m]      = acc0[v];
            yr[16 + m] = acc1[v];
            yr[32 + m] = acc2[v];
            yr[48 + m] = acc3[v];
        }
    }
}

// ---------------------------------------------------------------------------
// Zero a float buffer
// ---------------------------------------------------------------------------
__global__ void zero_kernel(float* p, long long n) {
    long long t = (long long)blockIdx.x * blockDim.x + threadIdx.x;
    if (t < n) p[t] = 0.0f;
}

// ---------------------------------------------------------------------------
// Edge scatter: acc[dst] += xt[src] * dinv[src]*dinv[dst]
// 4 threads per edge; each thread handles 16 contiguous floats (4x float4).
// ---------------------------------------------------------------------------
__global__ void edge_scatter_kernel(const long long* __restrict__ src,
                                    const long long* __restrict__ dst,
                                    const float* __restrict__ xt,
                                    const float* __restrict__ dinv,
                                    float* acc, long long nE) {
    long long t = (long long)blockIdx.x * blockDim.x + threadIdx.x;
    long long e = t >> 2;
    int seg = (int)(t & 3);
    if (e >= nE) return;
    long long s = src[e];
    long long d = dst[e];
    float w = dinv[s] * dinv[d];
    const float4* xs = (const float4*)(xt + s * 64 + seg * 16);
    float* ad = acc + d * 64 + seg * 16;
    #pragma unroll
    for (int q = 0; q < 4; ++q) {
        float4 v = xs[q];
        atomicAdd(ad + q * 4 + 0, v.x * w);
        atomicAdd(ad + q * 4 + 1, v.y * w);
        atomicAdd(ad + q * 4 + 2, v.z * w);
        atomicAdd(ad + q * 4 + 3, v.w * w);
    }
}

// ---------------------------------------------------------------------------
// Self-loop contribution + bias (+ optional ReLU), per (node, feature).
// Runs after edge_scatter (stream-serialized) -> no atomics needed.
// ---------------------------------------------------------------------------
__global__ void selfloop_bias_kernel(const float* __restrict__ xt,
                                     const float* __restrict__ dinv,
                                     const float* __restrict__ bias,
                                     float* acc, long long n_nodes, int do_relu) {
    long long t = (long long)blockIdx.x * blockDim.x + threadIdx.x;
    if (t >= n_nodes * 64) return;
    long long node = t >> 6;
    int f = (int)(t & 63);
    float di = dinv[node];
    float v = acc[t] + xt[t] * di * di + bias[f];
    if (do_relu) v = fmaxf(v, 0.0f);
    acc[t] = v;
}

// ---------------------------------------------------------------------------
// Launch
// ---------------------------------------------------------------------------
extern "C" void kernel_launch(void* const* d_in, const int* in_sizes, int n_in,
                              void* d_out, int out_size, void* d_ws, size_t ws_size,
                              hipStream_t stream) {
    const float*     x    = (const float*)d_in[0];
    const long long* eidx = (const long long*)d_in[1];
    const float*     W1   = (const float*)d_in[2];
    const float*     b1   = (const float*)d_in[3];
    const float*     W2   = (const float*)d_in[4];
    const float*     b2   = (const float*)d_in[5];

    const int N  = in_sizes[0] / 128;   // nodes
    const int E  = in_sizes[1] / 2;     // edges
    const int K1 = in_sizes[2] / 64;    // 128
    const int K2 = in_sizes[4] / 64;    // 64

    const long long* src = eidx;
    const long long* dst = eidx + E;

    float* ws_f = (float*)d_ws;
    float* dinv = ws_f;                              // [N]
    float* bufA = ws_f + N;                          // [N*64]  xt (transformed features)
    float* bufB = ws_f + N + (size_t)N * 64;         // [N*64]  accumulator -> h
    float* out  = (float*)d_out;                     // [N*64]

    const int B = 256;
    const long long NF = (long long)N * 64;

    // --- degree / normalization ---
    deg_init_kernel<<<(N + B - 1) / B, B, 0, stream>>>(dinv, N);
    deg_count_kernel<<<(E + B - 1) / B, B, 0, stream>>>(dst, dinv, E);
    deg_finalize_kernel<<<(N + B - 1) / B, B, 0, stream>>>(dinv, N);

    // --- layer 1: xt = x @ W1 (WMMA fp32), scatter, self-loop + bias + relu ---
    gemm_wmma_kernel<<<(N + 63) / 64, 128, 0, stream>>>(x, W1, bufA, N, K1);
    zero_kernel<<<(int)((NF + B - 1) / B), B, 0, stream>>>(bufB, NF);
    edge_scatter_kernel<<<(int)((4LL * E + B - 1) / B), B, 0, stream>>>(src, dst, bufA, dinv, bufB, E);
    selfloop_bias_kernel<<<(int)((NF + B - 1) / B), B, 0, stream>>>(bufA, dinv, b1, bufB, N, 1);

    // --- layer 2: xt2 = h @ W2 (WMMA fp32), scatter, self-loop + bias ---
    gemm_wmma_kernel<<<(N + 63) / 64, 128, 0, stream>>>(bufB, W2, bufA, N, K2);
    zero_kernel<<<(int)((NF + B - 1) / B), B, 0, stream>>>(out, NF);
    edge_scatter_kernel<<<(int)((4LL * E + B - 1) / B), B, 0, stream>>>(src, dst, bufA, dinv, out, E);
    selfloop_bias_kernel<<<(int)((NF + B - 1) / B), B, 0, stream>>>(bufA, dinv, b2, out, N, 0);
}